// Disentangler_52132313038898
// MI455X (gfx1250) — compile-verified
//
#include <hip/hip_runtime.h>
#include <hip/hip_bf16.h>

// Problem constants (from reference setup_inputs)
#define T_DIM 16
#define TOK 16384
#define E_DIM 128
#define N_NODE 8192
#define NUM_NODES 50000
#define COMP_LEN 64
#define MAX_LEN 782
#define COMP_DIM 32
#define OUT_D (COMP_LEN * COMP_DIM) // 2048

typedef __attribute__((ext_vector_type(2))) float v2f;
typedef __attribute__((ext_vector_type(8))) float v8f;

// ---------------------------------------------------------------- zero fill
__global__ __launch_bounds__(256) void k_zero(float* __restrict__ p, int n) {
    int i = blockIdx.x * 256 + threadIdx.x;
    if (i < n) p[i] = 0.0f;
}

// ------------------------------------------------- histogram of stacked_indices
// cnt layout: [NUM_NODES][COMP_LEN]  (so per-node counts for all 64 comps are contiguous)
__global__ __launch_bounds__(256) void k_hist(const int* __restrict__ si,
                                              float* __restrict__ cnt) {
    int i = blockIdx.x * 256 + threadIdx.x; // i = c*MAX_LEN + m
    if (i < COMP_LEN * MAX_LEN) {
        int c = i / MAX_LEN;
        int node = si[i];
        atomicAdd(&cnt[(size_t)node * COMP_LEN + c], 1.0f);
    }
}

// ---------------------------------------- fused LayerNorm(E=128) + group-of-4 sum
// One wave32 per row (t,k). Lane d loads x[4d..4d+3]; group d of the pooling is
// exactly lane d's 4 elements. Output Bm[t][k][d] pre-scaled by 1/(4*782).
__global__ __launch_bounds__(256) void k_ln_pool(const float* __restrict__ x,
                                                 const float* __restrict__ g,
                                                 const float* __restrict__ b,
                                                 float* __restrict__ Bm) {
    int row  = blockIdx.x * 8 + (threadIdx.x >> 5); // t*N_NODE + k, 0..131071
    int lane = threadIdx.x & 31;
    int t = row >> 13;       // / N_NODE
    int k = row & (N_NODE - 1);

    const float* xp = x + ((size_t)t * TOK + k) * E_DIM + lane * 4;
    float4 xv = *(const float4*)xp;

    float s = xv.x + xv.y + xv.z + xv.w;
    float q = xv.x * xv.x + xv.y * xv.y + xv.z * xv.z + xv.w * xv.w;
    #pragma unroll
    for (int off = 16; off; off >>= 1) {
        s += __shfl_xor(s, off, 32);
        q += __shfl_xor(q, off, 32);
    }
    float mu  = s * (1.0f / 128.0f);
    float var = q * (1.0f / 128.0f) - mu * mu;
    float rs  = rsqrtf(var + 1e-5f);

    float4 gv = *(const float4*)(g + lane * 4);
    float4 bv = *(const float4*)(b + lane * 4);
    float r = ((xv.x - mu) * rs * gv.x + bv.x) +
              ((xv.y - mu) * rs * gv.y + bv.y) +
              ((xv.z - mu) * rs * gv.z + bv.z) +
              ((xv.w - mu) * rs * gv.w + bv.w);

    Bm[(size_t)row * COMP_DIM + lane] = r * (1.0f / (4.0f * (float)MAX_LEN));
}

// ------------------------------------------------------------ WMMA f32 GEMM
// Per t: D(64x32) = A(64x8192) * B(8192x32), A[c][k] = cnt[node_idx[t,k]][c]
// gathered on the fly. One block (8 waves) per 16x16 D tile; waves split K,
// partial tiles reduced through LDS.
// V_WMMA_F32_16X16X4_F32 fragments (ISA 7.12.2):
//   A 16x4 f32: lane(l<16): M=l, v0=K0,v1=K1 ; lane>=16: K2,K3
//   B 4x16 f32: v0: row K0 (lanes0-15) / K2 (lanes16-31); v1: K1 / K3
//   -> both halves address kk = k0 + 2*half, kk+1.
__global__ __launch_bounds__(256) void k_gemm_wmma(const float* __restrict__ cnt,
                                                   const float* __restrict__ Bm,
                                                   const int* __restrict__ node_idx,
                                                   float* __restrict__ agg) {
    const int K = N_NODE;
    int blk = blockIdx.x;       // t*8 + tm*2 + tn
    int t  = blk >> 3;
    int tm = (blk >> 1) & 3;
    int tn = blk & 1;

    int wave = threadIdx.x >> 5;
    int lane = threadIdx.x & 31;
    int half = lane >> 4;
    int lid  = lane & 15;
    int c = tm * 16 + lid;   // A row  (comp index)
    int d = tn * 16 + lid;   // B col  (comp_dim index)

    const int*   nip = node_idx + (size_t)t * K;
    const float* bp  = Bm + (size_t)t * K * COMP_DIM + d;

    v8f acc = {};
    int kbeg = wave * (K / 8);
    int kend = kbeg + (K / 8);
    #pragma unroll 4
    for (int k = kbeg; k < kend; k += 4) {
        int kk = k + half * 2;
        int n0 = nip[kk];
        int n1 = nip[kk + 1];
        v2f a, bfrag;
        a.x = cnt[(size_t)n0 * COMP_LEN + c];
        a.y = cnt[(size_t)n1 * COMP_LEN + c];
        bfrag.x = bp[(size_t)kk * COMP_DIM];
        bfrag.y = bp[(size_t)(kk + 1) * COMP_DIM];
        acc = __builtin_amdgcn_wmma_f32_16x16x4_f32(
            /*neg_a=*/false, a, /*neg_b=*/false, bfrag,
            /*c_mod=*/(short)0, acc, /*reuse_a=*/false, /*reuse_b=*/false);
    }

    // Reduce the 8 per-wave partial tiles through LDS.
    // D fragment mapping: element (M = v + half*8, N = lid) lives in acc[v].
    __shared__ float red[8][256];
    #pragma unroll
    for (int v = 0; v < 8; ++v)
        red[wave][(v + half * 8) * 16 + lid] = acc[v];
    __syncthreads();

    float s = 0.0f;
    #pragma unroll
    for (int w = 0; w < 8; ++w) s += red[w][threadIdx.x];
    int m = threadIdx.x >> 4;
    int n = threadIdx.x & 15;
    agg[(size_t)t * OUT_D + (tm * 16 + m) * COMP_DIM + (tn * 16 + n)] = s;
}

// --------------------------------------------------------- final LayerNorm(2048)
__global__ __launch_bounds__(256) void k_final_ln(const float* __restrict__ agg,
                                                  const float* __restrict__ g,
                                                  const float* __restrict__ b,
                                                  float* __restrict__ out) {
    int t = blockIdx.x;
    const float* a = agg + (size_t)t * OUT_D;

    float s = 0.0f, q = 0.0f;
    for (int i = threadIdx.x; i < OUT_D; i += 256) {
        float v = a[i];
        s += v;
        q += v * v;
    }
    __shared__ float ss[8], qq[8];
    #pragma unroll
    for (int off = 16; off; off >>= 1) {
        s += __shfl_xor(s, off, 32);
        q += __shfl_xor(q, off, 32);
    }
    int wave = threadIdx.x >> 5, lane = threadIdx.x & 31;
    if (lane == 0) { ss[wave] = s; qq[wave] = q; }
    __syncthreads();
    if (wave == 0) {
        s = (lane < 8) ? ss[lane] : 0.0f;
        q = (lane < 8) ? qq[lane] : 0.0f;
        #pragma unroll
        for (int off = 4; off; off >>= 1) {
            s += __shfl_xor(s, off, 32);
            q += __shfl_xor(q, off, 32);
        }
        if (lane == 0) { ss[0] = s; qq[0] = q; }
    }
    __syncthreads();
    float mu  = ss[0] * (1.0f / (float)OUT_D);
    float var = qq[0] * (1.0f / (float)OUT_D) - mu * mu;
    float rs  = rsqrtf(var + 1e-5f);
    for (int i = threadIdx.x; i < OUT_D; i += 256)
        out[(size_t)t * OUT_D + i] = (a[i] - mu) * rs * g[i] + b[i];
}

extern "C" void kernel_launch(void* const* d_in, const int* in_sizes, int n_in,
                              void* d_out, int out_size, void* d_ws, size_t ws_size,
                              hipStream_t stream) {
    const float* x    = (const float*)d_in[0];
    const float* ln1g = (const float*)d_in[1];
    const float* ln1b = (const float*)d_in[2];
    const float* ln2g = (const float*)d_in[3];
    const float* ln2b = (const float*)d_in[4];
    const int* node_idx = (const int*)d_in[5];
    const int* si       = (const int*)d_in[6];
    float* out = (float*)d_out;

    // Workspace layout (all 256B-aligned):
    //   cnt : NUM_NODES*64 f32  = 12,800,000 B
    //   Bm  : 16*8192*32 f32    = 16,777,216 B
    //   agg : 16*2048 f32       =    131,072 B
    const size_t CNT_ELEMS = (size_t)NUM_NODES * COMP_LEN; // 3,200,000
    float* cnt = (float*)d_ws;
    float* Bm  = (float*)((char*)d_ws + 12800000);
    float* agg = (float*)((char*)d_ws + 12800000 + 16777216);

    // 1) zero histogram (must re-init every call; harness doesn't re-poison)
    k_zero<<<(int)((CNT_ELEMS + 255) / 256), 256, 0, stream>>>(cnt, (int)CNT_ELEMS);

    // 2) build cnt[node][c] from stacked_indices (64*782 atomic f32 adds)
    k_hist<<<(COMP_LEN * MAX_LEN + 255) / 256, 256, 0, stream>>>(si, cnt);

    // 3) fused LayerNorm + group-of-4 pooling -> B matrix (16,8192,32)
    k_ln_pool<<<(T_DIM * N_NODE) / 8, 256, 0, stream>>>(x, ln1g, ln1b, Bm);

    // 4) WMMA f32 GEMM: agg[t] = A[t](64x8192) @ B[t](8192x32)
    k_gemm_wmma<<<T_DIM * 8, 256, 0, stream>>>(cnt, Bm, node_idx, agg);

    // 5) final LayerNorm over 2048 per t
    k_final_ln<<<T_DIM, 256, 0, stream>>>(agg, ln2g, ln2b, out);
}